// GNNEncoder_47820165873981
// MI455X (gfx1250) — compile-verified
//
#include <hip/hip_runtime.h>

// GNN encoder for MI455X (gfx1250, wave32).
// - Dead-code eliminated: edge_proj + edge MLPs of layers 0/1 never reach the
//   output; only node pipeline (3 layers) + final-layer edge MLP are computed.
// - All GEMMs use V_WMMA_F32_16X16X4_F32 (exact fp32; workload is HBM/L2
//   bound, GEMM compute is negligible).
// - L2 (192MB) holds h ping-pong + agg + edge_index (~97MB): those accesses
//   stay regular-temporal. Pure streams (x read-once, 320MB e output) use
//   non-temporal hints so they don't evict the resident set.
// - Weights are staged in LDS pair-interleaved so each WMMA B-fragment is a
//   single aligned ds_load_b64 (no VGPR repacking).
// - Edge kernel preloads raw W via gfx1250 async global->LDS DMA
//   (global_load_async_to_lds_b128 + s_wait_asynccnt).

#define N_NODES 100000   // multiple of 16
#define N_EDGES 1250000  // multiple of 16
#define HDIM 64
#define LN_EPS 1e-5f

typedef __attribute__((ext_vector_type(2))) float v2f;
typedef __attribute__((ext_vector_type(4))) float v4f;
typedef __attribute__((ext_vector_type(8))) float v8f;

__device__ __forceinline__ v8f wmma4(v2f a, v2f b, v8f c) {
#if defined(__HIP_DEVICE_COMPILE__)
  return __builtin_amdgcn_wmma_f32_16x16x4_f32(
      false, a, false, b, (short)0, c, false, false);
#else
  return c;
#endif
}

// Write W(k, n0..n0+3) into pair-interleaved LDS layout:
//   sW2[(k>>1)*128 + 2n + (k&1)]  ->  fragment (kp,n) = one aligned float2.
__device__ __forceinline__ void stage_pairs(float* sW2, int i, v4f v) {
  const int k = i >> 4;           // 16 float4 per 64-wide row
  const int n0 = (i & 15) << 2;
  float* d = sW2 + (k >> 1) * 128 + (k & 1);
  d[(n0 + 0) * 2] = v.x;
  d[(n0 + 1) * 2] = v.y;
  d[(n0 + 2) * 2] = v.z;
  d[(n0 + 3) * 2] = v.w;
}

// ---------------------------------------------------------------------------
// h0 = x @ W(64x64) + b      one 16x64 tile per wave, K=64
// ---------------------------------------------------------------------------
__global__ __launch_bounds__(256) void k_node_proj(
    const float* __restrict__ x, const float* __restrict__ W,
    const float* __restrict__ bias, float* __restrict__ h) {
  __shared__ __align__(16) float sW2[64 * 64];  // 32 pairs x 64 cols x 2
  const int tid = threadIdx.x;
  for (int i = tid; i < (64 * 64) / 4; i += 256)
    stage_pairs(sW2, i, __builtin_nontemporal_load((const v4f*)W + i));
  __syncthreads();

  const int wave = tid >> 5, lane = tid & 31;
  const int mrow = lane & 15, khalf = lane >> 4;
  const int tile = blockIdx.x * 8 + wave;
  if (tile >= N_NODES / 16) return;  // no barriers below

  const float* arow = x + (size_t)(tile * 16 + mrow) * HDIM;
  v8f acc[4] = {};
#pragma unroll
  for (int t = 0; t < 16; ++t) {
    const int kp = 2 * t + khalf;  // k = 4t + 2*khalf (always even)
    v2f a = __builtin_nontemporal_load((const v2f*)(arow + 2 * kp));
#pragma unroll
    for (int nb = 0; nb < 4; ++nb) {
      v2f b = *(const v2f*)(sW2 + kp * 128 + (nb * 16 + mrow) * 2);
      acc[nb] = wmma4(a, b, acc[nb]);
    }
  }
#pragma unroll
  for (int nb = 0; nb < 4; ++nb) {
    const int n = nb * 16 + mrow;
    const float bn = bias[n];
#pragma unroll
    for (int r = 0; r < 8; ++r) {
      const int row = tile * 16 + r + 8 * khalf;
      h[(size_t)row * HDIM + n] = acc[nb][r] + bn;
    }
  }
}

// ---------------------------------------------------------------------------
__global__ __launch_bounds__(256) void k_zero(float* __restrict__ p, int n4) {
  int i = blockIdx.x * blockDim.x + threadIdx.x;
  if (i < n4) ((v4f*)p)[i] = v4f{0.f, 0.f, 0.f, 0.f};
}

// ---------------------------------------------------------------------------
// agg[row[e]] += h[col[e]]   4 threads/edge, native f32 atomics (L2-resident)
// ---------------------------------------------------------------------------
__global__ __launch_bounds__(256) void k_scatter(
    const long long* __restrict__ ei, const float* __restrict__ h,
    float* __restrict__ agg) {
  const long long tid = (long long)blockIdx.x * blockDim.x + threadIdx.x;
  if (tid >= (long long)N_EDGES * 4) return;
  const int e = (int)(tid >> 2), part = (int)(tid & 3);
  __builtin_prefetch(ei + e + 2048, 0, 0);  // index stream ahead
  const int r = (int)ei[e];
  const int c = (int)ei[N_EDGES + e];
  const v4f* src = (const v4f*)(h + (size_t)c * HDIM + part * 16);
  float* dst = agg + (size_t)r * HDIM + part * 16;
#pragma unroll
  for (int q = 0; q < 4; ++q) {
    v4f v = src[q];
    unsafeAtomicAdd(dst + q * 4 + 0, v.x);
    unsafeAtomicAdd(dst + q * 4 + 1, v.y);
    unsafeAtomicAdd(dst + q * 4 + 2, v.z);
    unsafeAtomicAdd(dst + q * 4 + 3, v.w);
  }
}

// ---------------------------------------------------------------------------
// hout = LN(relu([hin | agg] @ W(128x64) + b)) * g + be
// ---------------------------------------------------------------------------
__global__ __launch_bounds__(256) void k_node_mlp(
    const float* __restrict__ hin, const float* __restrict__ agg,
    const float* __restrict__ W, const float* __restrict__ bias,
    const float* __restrict__ g, const float* __restrict__ be,
    float* __restrict__ hout) {
  __shared__ __align__(16) float sW2[128 * 64];  // 64 pairs x 64 x 2
  __shared__ __align__(16) float sT[8][16 * 64];
  const int tid = threadIdx.x;
  for (int i = tid; i < (128 * 64) / 4; i += 256)
    stage_pairs(sW2, i, __builtin_nontemporal_load((const v4f*)W + i));
  __syncthreads();

  const int wave = tid >> 5, lane = tid & 31;
  const int mrow = lane & 15, khalf = lane >> 4;
  const int tile = blockIdx.x * 8 + wave;
  const bool valid = tile < N_NODES / 16;

  if (valid) {
    const float* a1 = hin + (size_t)(tile * 16 + mrow) * HDIM;
    const float* a2 = agg + (size_t)(tile * 16 + mrow) * HDIM;
    v8f acc[4] = {};
#pragma unroll
    for (int t = 0; t < 16; ++t) {  // K rows 0..63 <- hin
      const int kp = 2 * t + khalf;
      v2f a = *(const v2f*)(a1 + 2 * kp);
#pragma unroll
      for (int nb = 0; nb < 4; ++nb) {
        v2f b = *(const v2f*)(sW2 + kp * 128 + (nb * 16 + mrow) * 2);
        acc[nb] = wmma4(a, b, acc[nb]);
      }
    }
#pragma unroll
    for (int t = 0; t < 16; ++t) {  // K rows 64..127 <- agg
      const int kp = 2 * t + khalf;
      v2f a = *(const v2f*)(a2 + 2 * kp);
#pragma unroll
      for (int nb = 0; nb < 4; ++nb) {
        v2f b = *(const v2f*)(sW2 + (kp + 32) * 128 + (nb * 16 + mrow) * 2);
        acc[nb] = wmma4(a, b, acc[nb]);
      }
    }
    float* myT = sT[wave];
#pragma unroll
    for (int nb = 0; nb < 4; ++nb) {
      const int n = nb * 16 + mrow;
      const float bn = bias[n];
#pragma unroll
      for (int r = 0; r < 8; ++r)
        myT[(r + 8 * khalf) * 64 + n] = fmaxf(acc[nb][r] + bn, 0.f);
    }
  }
  __syncthreads();
  if (valid) {
    const float* trow = sT[wave] + mrow * 64 + khalf * 32;
    float vals[32], s = 0.f, s2 = 0.f;
#pragma unroll
    for (int i = 0; i < 32; ++i) {
      const float v = trow[i];
      vals[i] = v; s += v; s2 += v * v;
    }
    s  += __shfl_xor(s, 16, 32);
    s2 += __shfl_xor(s2, 16, 32);
    const float mean = s * (1.f / 64.f);
    const float var = s2 * (1.f / 64.f) - mean * mean;
    const float rstd = rsqrtf(var + LN_EPS);
    float* orow = hout + (size_t)(tile * 16 + mrow) * HDIM + khalf * 32;
#pragma unroll
    for (int q = 0; q < 8; ++q) {
      v4f o;
      const int n = khalf * 32 + q * 4;
      o.x = (vals[q * 4 + 0] - mean) * rstd * g[n + 0] + be[n + 0];
      o.y = (vals[q * 4 + 1] - mean) * rstd * g[n + 1] + be[n + 1];
      o.z = (vals[q * 4 + 2] - mean) * rstd * g[n + 2] + be[n + 2];
      o.w = (vals[q * 4 + 3] - mean) * rstd * g[n + 3] + be[n + 3];
      *((v4f*)orow + q) = o;  // h is L2-resident: regular temporal
    }
  }
}

// ---------------------------------------------------------------------------
// e = LN(relu([h2[row] | h2[col]] @ W(128x64) + b)) * g + be   (last layer)
// Raw W preloaded with gfx1250 async global->LDS DMA into the LN tile buffer
// (not yet live), then transposed LDS->LDS into pair-interleaved layout.
// ---------------------------------------------------------------------------
__global__ __launch_bounds__(256) void k_edge_mlp(
    const long long* __restrict__ ei, const float* __restrict__ h2,
    const float* __restrict__ W, const float* __restrict__ bias,
    const float* __restrict__ g, const float* __restrict__ be,
    float* __restrict__ eout) {
  __shared__ __align__(16) float sW2[128 * 64];
  __shared__ __align__(16) float sT[8][16 * 64];  // staging: raw W, later LN tiles
  const int tid = threadIdx.x;
  {
    const unsigned ldsBase = (unsigned)(uintptr_t)(&sT[0][0]);
#pragma unroll
    for (int it = 0; it < 8; ++it) {  // 8 x 256 lanes x 16B = 32KB
      const unsigned ldsAddr = ldsBase + (unsigned)((it * 256 + tid) * 16);
      const float* ga = W + (it * 256 + tid) * 4;
      asm volatile("global_load_async_to_lds_b128 %0, %1, off"
                   :: "v"(ldsAddr), "v"(ga) : "memory");
    }
    asm volatile("s_wait_asynccnt 0" ::: "memory");
  }
  __syncthreads();  // all waves' async copies done
  {
    const float* rawW = &sT[0][0];
    for (int i = tid; i < (128 * 64) / 4; i += 256)
      stage_pairs(sW2, i, *((const v4f*)rawW + i));
  }
  __syncthreads();  // sW2 ready; sT free for reuse

  const int wave = tid >> 5, lane = tid & 31;
  const int mrow = lane & 15, khalf = lane >> 4;
  const int tile = blockIdx.x * 8 + wave;
  const bool valid = tile < N_EDGES / 16;

  if (valid) {
    const int eIdx = tile * 16 + mrow;
    const int r = (int)ei[eIdx];
    const int c = (int)ei[N_EDGES + eIdx];
    const float* a1 = h2 + (size_t)r * HDIM;  // h2[row]
    const float* a2 = h2 + (size_t)c * HDIM;  // h2[col]
    v8f acc[4] = {};
#pragma unroll
    for (int t = 0; t < 16; ++t) {
      const int kp = 2 * t + khalf;
      v2f a = *(const v2f*)(a1 + 2 * kp);
#pragma unroll
      for (int nb = 0; nb < 4; ++nb) {
        v2f b = *(const v2f*)(sW2 + kp * 128 + (nb * 16 + mrow) * 2);
        acc[nb] = wmma4(a, b, acc[nb]);
      }
    }
#pragma unroll
    for (int t = 0; t < 16; ++t) {
      const int kp = 2 * t + khalf;
      v2f a = *(const v2f*)(a2 + 2 * kp);
#pragma unroll
      for (int nb = 0; nb < 4; ++nb) {
        v2f b = *(const v2f*)(sW2 + (kp + 32) * 128 + (nb * 16 + mrow) * 2);
        acc[nb] = wmma4(a, b, acc[nb]);
      }
    }
    float* myT = sT[wave];
#pragma unroll
    for (int nb = 0; nb < 4; ++nb) {
      const int n = nb * 16 + mrow;
      const float bn = bias[n];
#pragma unroll
      for (int r8 = 0; r8 < 8; ++r8)
        myT[(r8 + 8 * khalf) * 64 + n] = fmaxf(acc[nb][r8] + bn, 0.f);
    }
  }
  __syncthreads();
  if (valid) {
    const float* trow = sT[wave] + mrow * 64 + khalf * 32;
    float vals[32], s = 0.f, s2 = 0.f;
#pragma unroll
    for (int i = 0; i < 32; ++i) {
      const float v = trow[i];
      vals[i] = v; s += v; s2 += v * v;
    }
    s  += __shfl_xor(s, 16, 32);
    s2 += __shfl_xor(s2, 16, 32);
    const float mean = s * (1.f / 64.f);
    const float var = s2 * (1.f / 64.f) - mean * mean;
    const float rstd = rsqrtf(var + LN_EPS);
    float* orow = eout + (size_t)(tile * 16 + mrow) * HDIM + khalf * 32;
#pragma unroll
    for (int q = 0; q < 8; ++q) {
      v4f o;
      const int n = khalf * 32 + q * 4;
      o.x = (vals[q * 4 + 0] - mean) * rstd * g[n + 0] + be[n + 0];
      o.y = (vals[q * 4 + 1] - mean) * rstd * g[n + 1] + be[n + 1];
      o.z = (vals[q * 4 + 2] - mean) * rstd * g[n + 2] + be[n + 2];
      o.w = (vals[q * 4 + 3] - mean) * rstd * g[n + 3] + be[n + 3];
      // 320MB one-shot stream: non-temporal, keep L2 for h/agg/indices
      __builtin_nontemporal_store(o, (v4f*)orow + q);
    }
  }
}

// ---------------------------------------------------------------------------
extern "C" void kernel_launch(void* const* d_in, const int* in_sizes, int n_in,
                              void* d_out, int out_size, void* d_ws,
                              size_t ws_size, hipStream_t stream) {
  const float* x      = (const float*)d_in[0];
  const long long* ei = (const long long*)d_in[1];
  // d_in[2] edge_attr, d_in[5..6] edge_proj: dead code (e overwritten per layer)
  const float* npW  = (const float*)d_in[3];
  const float* npB  = (const float*)d_in[4];
  const float* nW   = (const float*)d_in[7];
  const float* nB   = (const float*)d_in[8];
  const float* nG   = (const float*)d_in[9];
  const float* nBe  = (const float*)d_in[10];
  const float* eWt  = (const float*)d_in[11];
  const float* eB   = (const float*)d_in[12];
  const float* eG   = (const float*)d_in[13];
  const float* eBe  = (const float*)d_in[14];

  float* out  = (float*)d_out;
  float* hB   = out;                          // h slot: N*64 floats
  float* eOut = out + (size_t)N_NODES * HDIM; // e slot: E*64 floats
  float* hA   = (float*)d_ws;                 // needs N*64*4 = 25.6 MB
  float* agg  = eOut;                         // reuse e region as agg scratch

  constexpr int NPROJ_BLOCKS = (N_NODES / 16 + 7) / 8;      // 782
  constexpr int ZERO_BLOCKS  = (N_NODES * HDIM / 4) / 256;  // 6250
  constexpr int SCAT_BLOCKS  = (N_EDGES * 4 + 255) / 256;   // 19532
  constexpr int EMLP_BLOCKS  = (N_EDGES / 16 + 7) / 8;      // 9766

  k_node_proj<<<NPROJ_BLOCKS, 256, 0, stream>>>(x, npW, npB, hA);

  const float* hcur = hA;  // ping-pong: h0=hA, h1=hB, h2=hA, h3=hB(=out)
  for (int l = 0; l < 3; ++l) {
    k_zero<<<ZERO_BLOCKS, 256, 0, stream>>>(agg, N_NODES * HDIM / 4);
    k_scatter<<<SCAT_BLOCKS, 256, 0, stream>>>(ei, hcur, agg);
    float* hnext = (l & 1) ? hA : hB;
    k_node_mlp<<<NPROJ_BLOCKS, 256, 0, stream>>>(
        hcur, agg, nW + (size_t)l * 128 * 64, nB + l * 64, nG + l * 64,
        nBe + l * 64, hnext);
    if (l == 2) {
      // reads h2 (= hcur); runs after node_mlp so agg region (== eOut) is dead
      k_edge_mlp<<<EMLP_BLOCKS, 256, 0, stream>>>(
          ei, hcur, eWt + (size_t)2 * 128 * 64, eB + 2 * 64, eG + 2 * 64,
          eBe + 2 * 64, eOut);
    }
    hcur = hnext;
  }
}